// SpikingCNN_1374389535092
// MI455X (gfx1250) — compile-verified
//
#include <hip/hip_runtime.h>

typedef __attribute__((ext_vector_type(2))) float v2f;
typedef __attribute__((ext_vector_type(8))) float v8f;

// CDNA5 async global->LDS copy path (gfx1250): direct memory->LDS, tracked by
// ASYNCcnt, no VGPR round-trip. Guarded so the file compiles on toolchains
// that don't expose the builtins (falls back to global_load + ds_store).
#if __has_builtin(__builtin_amdgcn_global_load_async_to_lds_b32) && \
    __has_builtin(__builtin_amdgcn_s_wait_asynccnt)
#define USE_ASYNC_LDS 1
#else
#define USE_ASYNC_LDS 0
#endif

typedef __attribute__((address_space(1))) int gint_t;
typedef __attribute__((address_space(3))) int lint_t;

// ---------------------------------------------------------------------------
// Problem constants (reference: B=64, Cin=2, T=65536)
//   stage1: conv(2->16,K=5,s=4,p=2) L=16384 -> pool 8192
//   stage2: conv(16->32,K=5,s=1,p=2) L=8192 -> pool 4096
//   stage3: conv(32->64,K=3,s=1,p=1) L=4096 -> pool 2048
// ---------------------------------------------------------------------------

__global__ void zero_stats(float* p) { p[threadIdx.x] = 0.f; }

// ---------------------------------------------------------------------------
// Stage-1 conv: direct (K-dim = Cin*K = 10, too skinny for WMMA).
// Each thread computes one (b, pos) for all 16 output channels; weights are
// uniform -> scalar loads. Accumulates per-channel sum/sumsq for BatchNorm.
// ---------------------------------------------------------------------------
__global__ void conv1_direct(const float* __restrict__ x,
                             const float* __restrict__ w,
                             const float* __restrict__ bias,
                             float* __restrict__ y,
                             float* __restrict__ gsum,
                             float* __restrict__ gsq) {
    constexpr int CIN = 2, K = 5, COUT = 16, T = 65536, L = 16384, STRIDE = 4, PAD = 2;
    __shared__ float s_sum[COUT], s_sq[COUT];
    int tid = threadIdx.x;
    if (tid < COUT) { s_sum[tid] = 0.f; s_sq[tid] = 0.f; }
    __syncthreads();

    int gidx = blockIdx.x * blockDim.x + tid;   // over B*L exactly
    int b = gidx / L;
    int p = gidx - b * L;
    const float* xb = x + (size_t)b * CIN * T;

    float acc[COUT];
#pragma unroll
    for (int co = 0; co < COUT; ++co) acc[co] = bias[co];

#pragma unroll
    for (int ci = 0; ci < CIN; ++ci) {
#pragma unroll
        for (int k = 0; k < K; ++k) {
            int ip = STRIDE * p + k - PAD;
            float xv = (ip >= 0 && ip < T) ? xb[(size_t)ci * T + ip] : 0.f;
#pragma unroll
            for (int co = 0; co < COUT; ++co)
                acc[co] = fmaf(w[co * CIN * K + ci * K + k], xv, acc[co]);
        }
    }

    float* yb = y + (size_t)b * COUT * L;
#pragma unroll
    for (int co = 0; co < COUT; ++co) {
        float v = acc[co];
        yb[(size_t)co * L + p] = v;
        atomicAdd(&s_sum[co], v);
        atomicAdd(&s_sq[co], v * v);
    }
    __syncthreads();
    if (tid < COUT) { atomicAdd(&gsum[tid], s_sum[tid]); atomicAdd(&gsq[tid], s_sq[tid]); }
}

// ---------------------------------------------------------------------------
// Stages 2/3: implicit-GEMM conv via V_WMMA_F32_16X16X4_F32 (exact f32).
//   D[16 couts x 16 pos] = sum_{chunks of 4 over Cin*K} A(16x4 W) * B(4x16 im2col)
// Input slab staged in LDS with the CDNA5 async global->LDS path (ASYNCcnt);
// 8 waves per WG cover CO_TILES x POS_TILES tiles; NPITER position-blocks per
// WG amortize weight-fragment loads. BN sum/sumsq fused.
// ---------------------------------------------------------------------------
template <int CIN, int KSZ, int COUT, int PAD, int POS_TILES, int NPITER, int LIN>
__global__ void conv_wmma(const float* __restrict__ x,
                          const float* __restrict__ w,
                          const float* __restrict__ bias,
                          float* __restrict__ y,
                          float* __restrict__ gsum,
                          float* __restrict__ gsq) {
    constexpr int KK = CIN * KSZ;           // 80 or 96
    constexpr int CHUNKS = KK / 4;          // 20 or 24
    constexpr int CO_TILES = COUT / 16;     // 2 or 4
    constexpr int WAVES = CO_TILES * POS_TILES;  // 8
    constexpr int THREADS = WAVES * 32;          // 256
    constexpr int POSW = POS_TILES * 16;    // 64 or 32
    constexpr int LW = POSW + KSZ - 1;      // slab width incl. halo
    constexpr int BLOCKS_PER_B = LIN / (POSW * NPITER);

    __shared__ float xs[CIN * LW];
    __shared__ float s_sum[COUT], s_sq[COUT];

    int tid  = threadIdx.x;
    int lane = tid & 31;
    int wave = tid >> 5;
    int n    = lane & 15;      // A row (cout) / B,D column (pos)
    int hi   = lane >> 4;      // half-wave selector
    int co_tile  = wave % CO_TILES;
    int pos_tile = wave / CO_TILES;
    int co_base  = co_tile * 16;

    if (tid < COUT) { s_sum[tid] = 0.f; s_sq[tid] = 0.f; }

    // --- A fragments: 16x4 weight tiles, ISA layout:
    //     lanes 0-15 hold M=0..15 {K0,K1}; lanes 16-31 hold M=0..15 {K2,K3}
    v2f afrag[CHUNKS];
    {
        int co_m = co_base + n;
#pragma unroll
        for (int c = 0; c < CHUNKS; ++c) {
            int kk = 4 * c + 2 * hi;
            afrag[c].x = w[co_m * KK + kk];
            afrag[c].y = w[co_m * KK + kk + 1];
        }
    }
    // --- B gather offsets into the LDS slab (kk = ci*KSZ + k flattening)
    int boff0[CHUNKS], boff1[CHUNKS];
#pragma unroll
    for (int c = 0; c < CHUNKS; ++c) {
        int kk0 = 4 * c + 2 * hi;
        int kk1 = kk0 + 1;
        boff0[c] = (kk0 / KSZ) * LW + (kk0 % KSZ);
        boff1[c] = (kk1 / KSZ) * LW + (kk1 % KSZ);
    }

    float biasr[8];
#pragma unroll
    for (int j = 0; j < 8; ++j) biasr[j] = bias[co_base + j + 8 * hi];

    int blk = blockIdx.x;
    int b   = blk / BLOCKS_PER_B;
    int p00 = (blk - b * BLOCKS_PER_B) * (POSW * NPITER);
    const float* xb = x + (size_t)b * CIN * LIN;
    float* yb = y + (size_t)b * COUT * LIN;

    float lsum[8], lsq[8];
#pragma unroll
    for (int j = 0; j < 8; ++j) { lsum[j] = 0.f; lsq[j] = 0.f; }

    for (int it = 0; it < NPITER; ++it) {
        int p0 = p00 + it * POSW;
        __syncthreads();
        // Cooperative slab load with halo. In-range lanes use the CDNA5 async
        // global->LDS copy (no VGPR round-trip, ASYNCcnt-tracked); out-of-range
        // halo lanes store zeros via ds_store.
        for (int idx = tid; idx < CIN * LW; idx += THREADS) {
            int ci = idx / LW;
            int j  = idx - ci * LW;
            int p  = p0 + j - PAD;
            if (p >= 0 && p < LIN) {
#if USE_ASYNC_LDS
                __builtin_amdgcn_global_load_async_to_lds_b32(
                    (gint_t*)(xb + (size_t)ci * LIN + p),
                    (lint_t*)(xs + idx),
                    0, 0);
#else
                xs[idx] = xb[(size_t)ci * LIN + p];
#endif
            } else {
                xs[idx] = 0.f;
            }
        }
#if USE_ASYNC_LDS
        __builtin_amdgcn_s_wait_asynccnt(0);
#endif
        __syncthreads();

        v8f acc = {};
        const float* xsp = xs + pos_tile * 16 + n;
#pragma unroll
        for (int c = 0; c < CHUNKS; ++c) {
            v2f bfrag;
            bfrag.x = xsp[boff0[c]];
            bfrag.y = xsp[boff1[c]];
            // D = A(16x4) * B(4x16) + C, exact f32 matrix op
            acc = __builtin_amdgcn_wmma_f32_16x16x4_f32(
                false, afrag[c], false, bfrag, (short)0, acc, false, false);
        }

        int pos = p0 + pos_tile * 16 + n;
#pragma unroll
        for (int j = 0; j < 8; ++j) {   // D: VGPR j -> row M=j+8*hi, col N=n
            float v = acc[j] + biasr[j];
            yb[(size_t)(co_base + j + 8 * hi) * LIN + pos] = v;
            lsum[j] += v;
            lsq[j]  += v * v;
        }
    }

    __syncthreads();
#pragma unroll
    for (int j = 0; j < 8; ++j) {
        atomicAdd(&s_sum[co_base + j + 8 * hi], lsum[j]);
        atomicAdd(&s_sq[co_base + j + 8 * hi], lsq[j]);
    }
    __syncthreads();
    if (tid < COUT) { atomicAdd(&gsum[tid], s_sum[tid]); atomicAdd(&gsq[tid], s_sq[tid]); }
}

// ---------------------------------------------------------------------------
// BatchNorm finalize: (sum, sumsq) -> per-channel (scale, shift)
// ---------------------------------------------------------------------------
__global__ void finalize_stats(const float* __restrict__ sum,
                               const float* __restrict__ sq,
                               const float* __restrict__ gamma,
                               const float* __restrict__ beta,
                               float* __restrict__ scale,
                               float* __restrict__ shift,
                               int C, float invN) {
    int c = threadIdx.x;
    if (c >= C) return;
    float mu  = sum[c] * invN;
    float var = sq[c] * invN - mu * mu;
    float r   = rsqrtf(var + 1e-5f);
    float s   = gamma[c] * r;
    scale[c] = s;
    shift[c] = fmaf(-mu, s, beta[c]);
}

// ---------------------------------------------------------------------------
// Fused BN-affine + LIF recurrence (subtract reset) + maxpool2.
// One thread streams one (b,c) row; float4 loads + prefetch hide latency
// across the inherently serial time recurrence.
// ---------------------------------------------------------------------------
__global__ void lif_scan_pool(const float* __restrict__ cur,
                              const float* __restrict__ scale,
                              const float* __restrict__ shift,
                              float* __restrict__ out,
                              int C, int L) {
    int row = blockIdx.x * blockDim.x + threadIdx.x;   // b*C + c
    int c = row % C;
    const float* in = cur + (size_t)row * L;
    float* op = out + (size_t)row * (L >> 1);
    float sc = scale[c], sh = shift[c];
    float mem = 0.f;
    for (int t = 0; t < L; t += 8) {
        if (t + 256 < L) __builtin_prefetch(in + t + 256, 0, 0);
        float4 v0 = *(const float4*)(in + t);
        float4 v1 = *(const float4*)(in + t + 4);
        float iv[8] = {v0.x, v0.y, v0.z, v0.w, v1.x, v1.y, v1.z, v1.w};
        float s[8];
#pragma unroll
        for (int u = 0; u < 8; ++u) {
            float i   = fmaf(sc, iv[u], sh);          // BN affine
            float rst = (mem > 1.0f) ? 1.0f : 0.0f;   // spike(mem_prev - thr)
            mem = fmaf(0.9f, mem, i) - rst;           // beta*mem + I - rst*thr
            s[u] = (mem > 1.0f) ? 1.0f : 0.0f;        // spike(mem - thr)
        }
        float4 o;
        o.x = fmaxf(s[0], s[1]);
        o.y = fmaxf(s[2], s[3]);
        o.z = fmaxf(s[4], s[5]);
        o.w = fmaxf(s[6], s[7]);
        *(float4*)(op + (t >> 1)) = o;
    }
}

// ---------------------------------------------------------------------------
// Driver. Workspace ping-pong:
//   A  : conv outputs (all stages are exactly 16,777,216 floats = 64 MB)
//   Bb : pooled spikes (stages 1/2 are exactly 8,388,608 floats = 32 MB)
//   stats: sum[64] | sq[64] | scale[64] | shift[64]
// ---------------------------------------------------------------------------
extern "C" void kernel_launch(void* const* d_in, const int* in_sizes, int n_in,
                              void* d_out, int out_size, void* d_ws, size_t ws_size,
                              hipStream_t stream) {
    const float* x   = (const float*)d_in[0];
    const float* w1  = (const float*)d_in[1];
    const float* b1  = (const float*)d_in[2];
    const float* g1  = (const float*)d_in[3];
    const float* be1 = (const float*)d_in[4];
    const float* w2  = (const float*)d_in[5];
    const float* b2  = (const float*)d_in[6];
    const float* g2  = (const float*)d_in[7];
    const float* be2 = (const float*)d_in[8];
    const float* w3  = (const float*)d_in[9];
    const float* b3  = (const float*)d_in[10];
    const float* g3  = (const float*)d_in[11];
    const float* be3 = (const float*)d_in[12];

    float* A    = (float*)d_ws;
    float* Bb   = A + 16777216;          // 64 MB in
    float* sum  = Bb + 8388608;          // +32 MB
    float* sq   = sum + 64;
    float* scl  = sq + 64;
    float* shf  = scl + 64;
    float* outp = (float*)d_out;

    // ---- stage 1: conv(2->16,s4) + BN-stats -> LIF+pool ----
    zero_stats<<<1, 128, 0, stream>>>(sum);
    conv1_direct<<<4096, 256, 0, stream>>>(x, w1, b1, A, sum, sq);
    finalize_stats<<<1, 64, 0, stream>>>(sum, sq, g1, be1, scl, shf, 16, 1.f / 1048576.f);
    lif_scan_pool<<<4, 256, 0, stream>>>(A, scl, shf, Bb, 16, 16384);

    // ---- stage 2: WMMA conv(16->32,K=5) ----
    zero_stats<<<1, 128, 0, stream>>>(sum);
    conv_wmma<16, 5, 32, 2, 4, 4, 8192><<<2048, 256, 0, stream>>>(Bb, w2, b2, A, sum, sq);
    finalize_stats<<<1, 64, 0, stream>>>(sum, sq, g2, be2, scl, shf, 32, 1.f / 524288.f);
    lif_scan_pool<<<8, 256, 0, stream>>>(A, scl, shf, Bb, 32, 8192);

    // ---- stage 3: WMMA conv(32->64,K=3) ----
    zero_stats<<<1, 128, 0, stream>>>(sum);
    conv_wmma<32, 3, 64, 1, 2, 4, 4096><<<2048, 256, 0, stream>>>(Bb, w3, b3, A, sum, sq);
    finalize_stats<<<1, 64, 0, stream>>>(sum, sq, g3, be3, scl, shf, 64, 1.f / 262144.f);
    lif_scan_pool<<<16, 256, 0, stream>>>(A, scl, shf, outp, 64, 4096);
}